// EnhancedGCN_32839319945349
// MI455X (gfx1250) — compile-verified
//
#include <hip/hip_runtime.h>
#include <hip/hip_bf16.h>

#define N_NODES 100000
#define DIM     128
#define N_EDGES 3200000
#define LN_EPS  1e-5f

typedef __attribute__((ext_vector_type(2))) float v2f;
typedef __attribute__((ext_vector_type(8))) float v8f;

// ---------------------------------------------------------------------------
// wave32 all-reduce (gfx1250 is wave32; use explicit width 32)
// ---------------------------------------------------------------------------
__device__ __forceinline__ float wave_sum32(float v) {
#pragma unroll
    for (int off = 16; off > 0; off >>= 1)
        v += __shfl_xor(v, off, 32);
    return v;
}

__device__ __forceinline__ void atom_add_f32(float* p, float v) {
    // hardware global_atomic_add_f32 (round-to-nearest-even, like the L2 fp adder)
    unsafeAtomicAdd(p, v);
}

// ---------------------------------------------------------------------------
// 1) zero agg accumulator (lives in d_out) + init deg with self-loop count
// ---------------------------------------------------------------------------
__global__ void k_init(float* __restrict__ agg, float* __restrict__ deg) {
    int i = blockIdx.x * 256 + threadIdx.x;
    if (i < N_NODES * DIM) agg[i] = 0.0f;
    if (i < N_NODES)       deg[i] = 1.0f;   // self loop
}

// ---------------------------------------------------------------------------
// 2) degree: +1 per edge destination
// ---------------------------------------------------------------------------
__global__ void k_degree(const int* __restrict__ ei, float* __restrict__ deg) {
    int e = blockIdx.x * 256 + threadIdx.x;
    if (e < N_EDGES) {
        int d = ei[N_EDGES + e];            // row 1 of (2,E) = dst
        atom_add_f32(&deg[d], 1.0f);
    }
}

// ---------------------------------------------------------------------------
// 3) deg -> rsqrt(deg)  (deg >= 1 always, matches reference's where-guard)
// ---------------------------------------------------------------------------
__global__ void k_dinv(float* __restrict__ deg) {
    int i = blockIdx.x * 256 + threadIdx.x;
    if (i < N_NODES) deg[i] = rsqrtf(deg[i]);
}

// ---------------------------------------------------------------------------
// 4) layernorm, one wave32 per row (32 lanes x float4 = 128)
// ---------------------------------------------------------------------------
__global__ void k_layernorm(const float* __restrict__ x,
                            const float* __restrict__ g,
                            const float* __restrict__ b,
                            float* __restrict__ out) {
    int row  = (blockIdx.x * blockDim.x + threadIdx.x) >> 5;
    int lane = threadIdx.x & 31;
    if (row >= N_NODES) return;
    float4 v = ((const float4*)(x + (size_t)row * DIM))[lane];
    float mu = wave_sum32(v.x + v.y + v.z + v.w) * (1.0f / DIM);
    float4 d = make_float4(v.x - mu, v.y - mu, v.z - mu, v.w - mu);
    float var = wave_sum32(d.x * d.x + d.y * d.y + d.z * d.z + d.w * d.w) * (1.0f / DIM);
    float rs = rsqrtf(var + LN_EPS);
    float4 gv = ((const float4*)g)[lane];
    float4 bv = ((const float4*)b)[lane];
    float4 o = make_float4(d.x * rs * gv.x + bv.x, d.y * rs * gv.y + bv.y,
                           d.z * rs * gv.z + bv.z, d.w * rs * gv.w + bv.w);
    ((float4*)(out + (size_t)row * DIM))[lane] = o;
}

// ---------------------------------------------------------------------------
// 5) GEMM: hw = hn @ W via V_WMMA_F32_16X16X4_F32.
//    Block = 256 thr (8 waves). Block owns 16 rows; wave w owns cols [16w,16w+16).
//    A tile (16x128) staged in LDS with pitch 132 floats to spread banks.
//    ISA 16x4 f32 A-frag layout: lanes 0-15 hold K{0,1}, lanes 16-31 hold K{2,3}.
// ---------------------------------------------------------------------------
__global__ __launch_bounds__(256) void k_gemm(const float* __restrict__ A,
                                              const float* __restrict__ W,
                                              float* __restrict__ C) {
    __shared__ float sA[16 * 132];
    const int row0 = blockIdx.x * 16;
    const int t = threadIdx.x;

    // cooperative A-tile load: 512 float4s over 256 threads
#pragma unroll
    for (int f = t; f < 512; f += 256) {
        int r = f >> 5, c4 = f & 31;
        float4 v = ((const float4*)(A + (size_t)(row0 + r) * DIM))[c4];
        ((float4*)(sA + r * 132))[c4] = v;
    }
    __syncthreads();

    const int wv = t >> 5;
    const int lane = t & 31;
    const int n  = (wv << 4) + (lane & 15);   // output column
    const int m  = lane & 15;                 // A-frag row
    const int kb = (lane >> 4) << 1;          // K sub-offset: 0 or 2

    const float* arow = sA + m * 132;
    v8f acc = {};
#pragma unroll
    for (int k0 = 0; k0 < DIM; k0 += 4) {
        v2f a, b;
        a.x = arow[k0 + kb];
        a.y = arow[k0 + kb + 1];
        b.x = W[(size_t)(k0 + kb)     * DIM + n];
        b.y = W[(size_t)(k0 + kb + 1) * DIM + n];
        acc = __builtin_amdgcn_wmma_f32_16x16x4_f32(
            /*neg_a=*/false, a, /*neg_b=*/false, b,
            /*c_mod=*/(short)0, acc, /*reuse_a=*/false, /*reuse_b=*/false);
    }

    // C/D layout: VGPR r -> M = r + 8*(lane/16), N = lane%16
    const int mh = (lane >> 4) << 3;
    float* crow = C + (size_t)row0 * DIM + n;
#pragma unroll
    for (int r = 0; r < 8; ++r)
        crow[(size_t)(mh + r) * DIM] = acc[r];
}

// ---------------------------------------------------------------------------
// 6) edge scatter: one wave per edge; lane handles a float4 chunk of the row.
// ---------------------------------------------------------------------------
__global__ void k_scatter(const int* __restrict__ ei,
                          const float* __restrict__ hw,
                          const float* __restrict__ dinv,
                          float* __restrict__ agg) {
    int e    = (blockIdx.x * blockDim.x + threadIdx.x) >> 5;
    int lane = threadIdx.x & 31;
    if (e >= N_EDGES) return;
    __builtin_prefetch(ei + e + 2048, 0, 1);               // global_prefetch_b8
    __builtin_prefetch(ei + N_EDGES + e + 2048, 0, 1);
    int s = ei[e];
    int d = ei[N_EDGES + e];
    float nrm = dinv[s] * dinv[d];
    float4 v = ((const float4*)(hw + (size_t)s * DIM))[lane];
    float* dp = agg + (size_t)d * DIM + (lane << 2);
    atom_add_f32(dp + 0, v.x * nrm);
    atom_add_f32(dp + 1, v.y * nrm);
    atom_add_f32(dp + 2, v.z * nrm);
    atom_add_f32(dp + 3, v.w * nrm);
}

// ---------------------------------------------------------------------------
// 7) final: t = x + agg + hw*dinv^2 (self loop) + bias; out = LN(t, g2, b2)
//    agg lives in `out`; wave reads its row fully before writing it.
// ---------------------------------------------------------------------------
__global__ void k_final(const float* __restrict__ x,
                        const float* __restrict__ hw,
                        const float* __restrict__ dinv,
                        const float* __restrict__ bias,
                        const float* __restrict__ g2,
                        const float* __restrict__ b2,
                        float* __restrict__ out) {
    int row  = (blockIdx.x * blockDim.x + threadIdx.x) >> 5;
    int lane = threadIdx.x & 31;
    if (row >= N_NODES) return;
    float di = dinv[row];
    float self = di * di;
    float4 av = ((const float4*)(out + (size_t)row * DIM))[lane];
    float4 hv = ((const float4*)(hw  + (size_t)row * DIM))[lane];
    float4 xv = ((const float4*)(x   + (size_t)row * DIM))[lane];
    float4 bv = ((const float4*)bias)[lane];
    float4 tv = make_float4(xv.x + av.x + hv.x * self + bv.x,
                            xv.y + av.y + hv.y * self + bv.y,
                            xv.z + av.z + hv.z * self + bv.z,
                            xv.w + av.w + hv.w * self + bv.w);
    float mu = wave_sum32(tv.x + tv.y + tv.z + tv.w) * (1.0f / DIM);
    float4 dv = make_float4(tv.x - mu, tv.y - mu, tv.z - mu, tv.w - mu);
    float var = wave_sum32(dv.x * dv.x + dv.y * dv.y + dv.z * dv.z + dv.w * dv.w) * (1.0f / DIM);
    float rs = rsqrtf(var + LN_EPS);
    float4 gv = ((const float4*)g2)[lane];
    float4 b2v = ((const float4*)b2)[lane];
    float4 o = make_float4(dv.x * rs * gv.x + b2v.x, dv.y * rs * gv.y + b2v.y,
                           dv.z * rs * gv.z + b2v.z, dv.w * rs * gv.w + b2v.w);
    ((float4*)(out + (size_t)row * DIM))[lane] = o;
}

// ---------------------------------------------------------------------------
extern "C" void kernel_launch(void* const* d_in, const int* in_sizes, int n_in,
                              void* d_out, int out_size, void* d_ws, size_t ws_size,
                              hipStream_t stream) {
    const float* x    = (const float*)d_in[0];
    const int*   ei   = (const int*)  d_in[1];   // (2, E): [0..E)=src, [E..2E)=dst
    const float* W    = (const float*)d_in[2];
    const float* bias = (const float*)d_in[3];
    const float* g1   = (const float*)d_in[4];
    const float* b1   = (const float*)d_in[5];
    const float* g2   = (const float*)d_in[6];
    const float* b2   = (const float*)d_in[7];
    float* out = (float*)d_out;

    float* hn  = (float*)d_ws;                       // N*D  layernormed x
    float* hw  = hn + (size_t)N_NODES * DIM;         // N*D  hn @ W
    float* deg = hw + (size_t)N_NODES * DIM;         // N    degree -> dinv

    k_init     <<<(N_NODES * DIM + 255) / 256, 256, 0, stream>>>(out, deg);
    k_degree   <<<(N_EDGES + 255) / 256,       256, 0, stream>>>(ei, deg);
    k_dinv     <<<(N_NODES + 255) / 256,       256, 0, stream>>>(deg);
    k_layernorm<<<(N_NODES + 7) / 8,           256, 0, stream>>>(x, g1, b1, hn);
    k_gemm     <<< N_NODES / 16,               256, 0, stream>>>(hn, W, hw);
    k_scatter  <<<(N_EDGES + 7) / 8,           256, 0, stream>>>(ei, hw, deg, out);
    k_final    <<<(N_NODES + 7) / 8,           256, 0, stream>>>(x, hw, deg, bias, g2, b2, out);
}